// NoiseProjector_59837484368482
// MI455X (gfx1250) — compile-verified
//
#include <hip/hip_runtime.h>
#include <hip/hip_bf16.h>

// ---------------------------------------------------------------------------
// MI455X (gfx1250) implementation.
//
// Dominant cost: third-order head  [64, 262144] @ Wt^T[262144, 512]
//   -> stream 537 MB of Wt once (HBM roofline ~23 us), A generated on the fly
//      from fc (16 KB) since outer3[b, i*4096 + j*64 + l] = fc[b,i]*fc[b,j]*fc[b,l].
//   -> fp32 WMMA (V_WMMA_F32_16X16X4_F32), wave32, 16x16 C tiles.
//   -> Wt staged through LDS via GLOBAL_LOAD_ASYNC_TO_LDS_B128 (ASYNCcnt).
// cov head is folded in as chunk #64 (same A pattern with scale 1, B = Wc).
// mean head + biases + chunk reduction done in a deterministic epilogue kernel.
// ---------------------------------------------------------------------------

typedef __attribute__((ext_vector_type(2))) float fx2;
typedef __attribute__((ext_vector_type(8))) float fx8;
typedef int npv4i __attribute__((vector_size(16)));   // matches builtin proto

#define B_SZ   64
#define C_IN   3
#define H_IN   224
#define C1     32
#define H1DIM  112
#define FDIM   64
#define H2DIM  56
#define ODIM   512

#define KTOT   (FDIM * FDIM * FDIM)   // 262144
#define KCHUNK (FDIM * FDIM)          // 4096 per chunk (fixed i)
#define NCHUNK (FDIM + 1)             // 64 third-order chunks + 1 cov chunk
#define NTILES (ODIM / 16)            // 32
#define KSTAGE 256                    // k-elements staged through LDS
#define NSTAGES (KCHUNK / KSTAGE)     // 16

#define FCPITCH 66                    // padded LDS pitch for fc table
#define BPITCH  260                   // padded LDS pitch for B tile rows

// ---- CDNA5 async global->LDS copy (ASYNCcnt-tracked, bypasses VGPRs) ------
#if defined(__has_builtin)
#if __has_builtin(__builtin_amdgcn_global_load_async_to_lds_b128)
#define HAVE_ASYNC_LDS 1
#endif
#endif

#ifdef HAVE_ASYNC_LDS
__device__ static inline void np_async_copy_b128(const float* gsrc,
                                                 float* ldst) {
  // AS1 via inttoptr (64-bit); AS3 via inttoptr of the low 32 bits, which on
  // amdgcn are exactly the LDS byte offset of a generic shared pointer.
  __builtin_amdgcn_global_load_async_to_lds_b128(
      (__attribute__((address_space(1))) npv4i*)(unsigned long long)gsrc,
      (__attribute__((address_space(3))) npv4i*)(unsigned int)(unsigned long long)
          ldst,
      0, 0);
}
__device__ static inline void np_wait_async() {
#if __has_builtin(__builtin_amdgcn_s_wait_asynccnt)
  __builtin_amdgcn_s_wait_asynccnt(0);
#else
  asm volatile("s_wait_asynccnt 0" ::: "memory");
#endif
}
#endif

// ---------------------------------------------------------------------------
// conv1: [64,3,224,224] -> relu -> [64,32,112,112]   (stride 2, pad 1)
// ---------------------------------------------------------------------------
__global__ __launch_bounds__(256) void np_conv1_kernel(
    const float* __restrict__ x, const float* __restrict__ W1,
    const float* __restrict__ b1, float* __restrict__ h1) {
  int idx = blockIdx.x * 256 + threadIdx.x;
  const int total = B_SZ * C1 * H1DIM * H1DIM;
  if (idx >= total) return;
  int ow = idx % H1DIM;
  int t = idx / H1DIM;
  int oh = t % H1DIM;
  t /= H1DIM;
  int oc = t % C1;
  int b = t / C1;

  float acc = b1[oc];
#pragma unroll
  for (int ic = 0; ic < C_IN; ++ic) {
#pragma unroll
    for (int r = 0; r < 3; ++r) {
      int ih = oh * 2 - 1 + r;
      if (ih < 0 || ih >= H_IN) continue;
#pragma unroll
      for (int s = 0; s < 3; ++s) {
        int iw = ow * 2 - 1 + s;
        if (iw < 0 || iw >= H_IN) continue;
        acc += x[((b * C_IN + ic) * H_IN + ih) * H_IN + iw] *
               W1[((oc * C_IN + ic) * 3 + r) * 3 + s];
      }
    }
  }
  h1[idx] = fmaxf(acc, 0.0f);
}

// ---------------------------------------------------------------------------
// conv2 + relu + global average pool, fused:  h1 -> feat[64,64]
// grid = (8 oc-tiles, 64 batch), block = 256.  Deterministic tree reduction.
// ---------------------------------------------------------------------------
__global__ __launch_bounds__(256) void np_conv2_pool_kernel(
    const float* __restrict__ h1, const float* __restrict__ W2,
    const float* __restrict__ b2, float* __restrict__ feat) {
  const int b = blockIdx.y;
  const int oc0 = blockIdx.x * 8;
  const int tid = threadIdx.x;

  __shared__ float W2s[8 * C1 * 9];   // 2304 floats
  __shared__ float red[256];

  for (int i = tid; i < 8 * C1 * 9; i += 256) W2s[i] = W2[oc0 * C1 * 9 + i];
  __syncthreads();

  float b2r[8];
#pragma unroll
  for (int i = 0; i < 8; ++i) b2r[i] = b2[oc0 + i];

  float pooled[8];
#pragma unroll
  for (int i = 0; i < 8; ++i) pooled[i] = 0.0f;

  for (int pos = tid; pos < H2DIM * H2DIM; pos += 256) {
    int oh = pos / H2DIM, ow = pos % H2DIM;
    float s8[8];
#pragma unroll
    for (int i = 0; i < 8; ++i) s8[i] = 0.0f;

    for (int ic = 0; ic < C1; ++ic) {
#pragma unroll
      for (int r = 0; r < 3; ++r) {
        int ih = oh * 2 - 1 + r;
        if (ih < 0 || ih >= H1DIM) continue;
#pragma unroll
        for (int s = 0; s < 3; ++s) {
          int iw = ow * 2 - 1 + s;
          if (iw < 0 || iw >= H1DIM) continue;
          float v = h1[((b * C1 + ic) * H1DIM + ih) * H1DIM + iw];
          int wb = ic * 9 + r * 3 + s;
#pragma unroll
          for (int i = 0; i < 8; ++i) s8[i] += v * W2s[i * C1 * 9 + wb];
        }
      }
    }
#pragma unroll
    for (int i = 0; i < 8; ++i) pooled[i] += fmaxf(s8[i] + b2r[i], 0.0f);
  }

  for (int i = 0; i < 8; ++i) {
    red[tid] = pooled[i];
    __syncthreads();
    for (int s = 128; s > 0; s >>= 1) {
      if (tid < s) red[tid] += red[tid + s];
      __syncthreads();
    }
    if (tid == 0)
      feat[b * FDIM + oc0 + i] = red[0] * (1.0f / (H2DIM * H2DIM));
    __syncthreads();
  }
}

// ---------------------------------------------------------------------------
// fc = feat - mean(feat, axis=1)    grid=64, block=64
// ---------------------------------------------------------------------------
__global__ __launch_bounds__(64) void np_center_kernel(
    const float* __restrict__ feat, float* __restrict__ fc) {
  int b = blockIdx.x;
  int f = threadIdx.x;
  __shared__ float red[64];
  float v = feat[b * FDIM + f];
  red[f] = v;
  __syncthreads();
  for (int s = 32; s > 0; s >>= 1) {
    if (f < s) red[f] += red[f + s];
    __syncthreads();
  }
  float mean = red[0] * (1.0f / FDIM);
  fc[b * FDIM + f] = v - mean;
}

// ---------------------------------------------------------------------------
// Big fused GEMM: partials[ntile][chunk][64][16] =
//   (on-the-fly A over K-chunk) x (Wt or Wc tile), fp32 WMMA 16x16x4.
// grid = (32 n-tiles, 65 chunks), block = 128 (4 waves, one 16x16 C-tile each)
// A-operand fc pairs live in registers (fv[16]); only B comes from LDS in the
// hot loop.  B tiles staged via async global->LDS copies when available.
// ---------------------------------------------------------------------------
__global__ __launch_bounds__(128) void np_third_gemm_kernel(
    const float* __restrict__ fcp, const float* __restrict__ Wt,
    const float* __restrict__ Wc, float* __restrict__ part) {
  const int ntile = blockIdx.x;   // 0..31  (output cols o = 16*ntile .. +15)
  const int chunk = blockIdx.y;   // 0..63 third-order, 64 = cov chunk
  const int tid = threadIdx.x;
  const int wave = tid >> 5;
  const int lane = tid & 31;

  __shared__ float fcS[FDIM * FCPITCH];       // raw fc table, padded
  __shared__ float Bs[16 * BPITCH];           // B tile [o][k] padded pitch

  for (int i = tid; i < FDIM * FDIM; i += 128) {
    int mm = i >> 6, ll = i & 63;
    fcS[mm * FCPITCH + ll] = fcp[i];
  }
  __syncthreads();

  const bool isCov = (chunk == FDIM);
  const float* Bsrc = isCov ? Wc : Wt;
  const size_t Krow = isCov ? (size_t)(FDIM * FDIM) : (size_t)KTOT;
  const size_t kbase = isCov ? 0 : (size_t)chunk * KCHUNK;

  const int m = wave * 16 + (lane & 15);     // batch row this lane serves
  const int n = lane & 15;                   // output col within tile
  const int koff = (lane >= 16) ? 2 : 0;     // ISA A/B K-slot split

  const float s_m = isCov ? 1.0f : fcS[m * FCPITCH + chunk];

  // Per-lane A-operand fc pairs: fv[i] = fc[m][4i+koff .. 4i+koff+1].
  // Constant across the whole chunk -> keep in VGPRs.
  float2 fv[16];
#pragma unroll
  for (int i = 0; i < 16; ++i)
    fv[i] = *(const float2*)&fcS[m * FCPITCH + i * 4 + koff];

  fx8 c = {0.f, 0.f, 0.f, 0.f, 0.f, 0.f, 0.f, 0.f};

  const float* grow = Bsrc + (size_t)(ntile * 16) * Krow + kbase;

  for (int stage = 0; stage < NSTAGES; ++stage) {
    __syncthreads();
    // Stage 16 rows x 256 k of B into LDS (pitch-padded layout).
#ifdef HAVE_ASYNC_LDS
#pragma unroll
    for (int i = 0; i < 8; ++i) {
      int q = tid + i * 128;                 // 0..1023
      int o = q >> 6;                        // row 0..15
      int c4 = q & 63;                       // float4 within row
      np_async_copy_b128(grow + (size_t)o * Krow + stage * KSTAGE + c4 * 4,
                         &Bs[o * BPITCH + c4 * 4]);
      if (stage + 1 < NSTAGES) {
        __builtin_prefetch(
            (const void*)(grow + (size_t)o * Krow + (stage + 1) * KSTAGE +
                          c4 * 4), 0, 3);
      }
    }
    np_wait_async();
#else
#pragma unroll
    for (int i = 0; i < 8; ++i) {
      int q = tid + i * 128;
      int o = q >> 6;
      int c4 = q & 63;
      const float4* src =
          (const float4*)(grow + (size_t)o * Krow + stage * KSTAGE) + c4;
      float4 v = *src;
      *(float4*)&Bs[o * BPITCH + c4 * 4] = v;
      if (stage + 1 < NSTAGES) {
        __builtin_prefetch(
            (const void*)((const float4*)(grow + (size_t)o * Krow +
                                          (stage + 1) * KSTAGE) + c4), 0, 3);
      }
    }
#endif
    __syncthreads();

    // Compute: j spans [stage*4, stage*4+3], l spans 0..63.
#pragma unroll
    for (int j4 = 0; j4 < 4; ++j4) {
      int j = stage * 4 + j4;
      float u = s_m * fcS[m * FCPITCH + j];
#pragma unroll
      for (int l = 0; l < 64; l += 4) {
        // A[m, k] = u * fc[m, l'] ; lanes 0-15 hold K=0,1; 16-31 hold K=2,3.
        fx2 a;
        a.x = u * fv[l >> 2].x;
        a.y = u * fv[l >> 2].y;
        // B[k, n] from LDS, n = lane&15, same K-slot split.
        float2 bv = *(const float2*)&Bs[n * BPITCH + j4 * 64 + l + koff];
        fx2 bb;
        bb.x = bv.x;
        bb.y = bv.y;
        c = __builtin_amdgcn_wmma_f32_16x16x4_f32(
            /*neg_a=*/false, a, /*neg_b=*/false, bb,
            /*c_mod=*/(short)0, c, /*reuse_a=*/false, /*reuse_b=*/false);
      }
    }
  }

  // D layout: VGPR r -> (M = r + 8*(lane>=16), N = lane&15)
  float* pout = part + (size_t)(ntile * NCHUNK + chunk) * (64 * 16);
  int mbase = wave * 16 + ((lane >= 16) ? 8 : 0);
#pragma unroll
  for (int r = 0; r < 8; ++r) pout[(mbase + r) * 16 + n] = c[r];
}

// ---------------------------------------------------------------------------
// Epilogue: out[b,o] = bm+bc+bt + feat@Wm^T + sum over 65 chunk partials.
// Fixed-order reduction -> deterministic.
// ---------------------------------------------------------------------------
__global__ __launch_bounds__(256) void np_reduce_kernel(
    const float* __restrict__ part, const float* __restrict__ feat,
    const float* __restrict__ Wm, const float* __restrict__ bm,
    const float* __restrict__ bc, const float* __restrict__ bt,
    float* __restrict__ out) {
  int idx = blockIdx.x * 256 + threadIdx.x;
  if (idx >= B_SZ * ODIM) return;
  int o = idx % ODIM;
  int b = idx / ODIM;

  float acc = bm[o] + bc[o] + bt[o];
  const float* wm = Wm + o * FDIM;
  const float* fb = feat + b * FDIM;
#pragma unroll
  for (int f = 0; f < FDIM; ++f) acc += fb[f] * wm[f];

  int ntile = o >> 4, n = o & 15;
  for (int ch = 0; ch < NCHUNK; ++ch)
    acc += part[((size_t)(ntile * NCHUNK + ch) * 64 + b) * 16 + n];

  out[idx] = acc;
}

// ---------------------------------------------------------------------------
extern "C" void kernel_launch(void* const* d_in, const int* in_sizes, int n_in,
                              void* d_out, int out_size, void* d_ws,
                              size_t ws_size, hipStream_t stream) {
  (void)in_sizes; (void)n_in; (void)out_size; (void)ws_size;
  const float* x  = (const float*)d_in[0];
  const float* W1 = (const float*)d_in[1];
  const float* b1 = (const float*)d_in[2];
  const float* W2 = (const float*)d_in[3];
  const float* b2 = (const float*)d_in[4];
  const float* Wm = (const float*)d_in[5];
  const float* bm = (const float*)d_in[6];
  const float* Wc = (const float*)d_in[7];
  const float* bc = (const float*)d_in[8];
  const float* Wt = (const float*)d_in[9];
  const float* bt = (const float*)d_in[10];
  float* out = (float*)d_out;

  // Workspace carve-up (floats): h1 | feat | fc | partials
  float* h1   = (float*)d_ws;                              // 25,690,112
  float* feat = h1 + (size_t)B_SZ * C1 * H1DIM * H1DIM;    // 4,096
  float* fc   = feat + (size_t)B_SZ * FDIM;                // 4,096
  float* part = fc + (size_t)B_SZ * FDIM;                  // 32*65*64*16

  const int conv1_total = B_SZ * C1 * H1DIM * H1DIM;
  np_conv1_kernel<<<(conv1_total + 255) / 256, 256, 0, stream>>>(x, W1, b1, h1);

  np_conv2_pool_kernel<<<dim3(FDIM / 8, B_SZ), 256, 0, stream>>>(h1, W2, b2,
                                                                 feat);

  np_center_kernel<<<B_SZ, FDIM, 0, stream>>>(feat, fc);

  np_third_gemm_kernel<<<dim3(NTILES, NCHUNK), 128, 0, stream>>>(fc, Wt, Wc,
                                                                 part);

  np_reduce_kernel<<<(B_SZ * ODIM + 255) / 256, 256, 0, stream>>>(
      part, feat, Wm, bm, bc, bt, out);
}